// SharedVN_DGCNNLayer_86792699118050
// MI455X (gfx1250) — compile-verified
//
#include <hip/hip_runtime.h>
#include <math.h>

typedef float v2f __attribute__((ext_vector_type(2)));
typedef float v8f __attribute__((ext_vector_type(8)));

#define VN_EPS   1e-6f
#define BN_EPS   1e-5f
#define KNN      20
#define NEG_SLOPE 0.2f
#define NB       8       // batch
#define NPTS     2048    // points
#define NO       21      // output channels

// ---------------------------------------------------------------------------
// K0: zero the stats accumulators (must run every launch; ws is not re-poisoned)
// ---------------------------------------------------------------------------
__global__ void zero_stats_kernel(float* __restrict__ sums) {
    int t = threadIdx.x;
    if (t < 128) sums[t] = 0.0f;
}

// ---------------------------------------------------------------------------
// K1: KNN. One block = one batch x 16 rows. 256 threads = 8 waves.
// Gram tile via V_WMMA_F32_16X16X4_F32 (K=4: 3 coords + zero-padded 4th row
// staged in LDS so fragment loads are branch-free ds_loads, keeping EXEC
// uniform around the WMMA).
// pdist = 2*inner - xx_row - xx_col ; per-row top-20 kept in registers,
// merged through LDS.
// ---------------------------------------------------------------------------
__global__ void knn_kernel(const float* __restrict__ x, int* __restrict__ idxOut) {
    const int bi      = blockIdx.x / (NPTS / 16);
    const int rowTile = blockIdx.x % (NPTS / 16);
    const int rowBase = rowTile * 16;
    const float* xb = x + (size_t)bi * 3 * NPTS;

    extern __shared__ char smem[];
    float* xs   = (float*)smem;                       // [4][NPTS], row 3 = 0
    float* xx   = xs + 4 * NPTS;                      // [NPTS]
    float* dist = xx + NPTS;                          // [16][264]
    float* mVal = dist + 16 * 264;                    // [16][16][KNN]
    int*   mIdx = (int*)(mVal + 16 * 16 * KNN);       // [16][16][KNN]

    const int t = threadIdx.x;

    // stage coordinates (4th row zeroed for K=4 padding) + squared norms
    for (int c = t; c < NPTS; c += 256) {
        float x0 = xb[c], x1 = xb[NPTS + c], x2 = xb[2 * NPTS + c];
        xs[c]            = x0;
        xs[NPTS + c]     = x1;
        xs[2 * NPTS + c] = x2;
        xs[3 * NPTS + c] = 0.0f;
        xx[c] = x0 * x0 + x1 * x1 + x2 * x2;
    }
    __syncthreads();

    const int lane = t & 31;
    const int wave = t >> 5;
    const int mn   = lane & 15;        // M for A-frag, N for B-frag
    const int kb   = (lane >> 4) * 2;  // K pair base: 0 or 2
    const int hi   = (lane >> 4) & 1;

    // A fragment: 16 rows x 4 coords (branch-free LDS reads)
    v2f afrag;
    afrag.x = xs[kb * NPTS + rowBase + mn];
    afrag.y = xs[(kb + 1) * NPTS + rowBase + mn];

    // loop-invariant row norms for this lane's 8 D-matrix rows
    float xxr[8];
#pragma unroll
    for (int r = 0; r < 8; ++r) xxr[r] = xx[rowBase + hi * 8 + r];

    // per-thread top-k state: thread handles row (t&15), scan slot (t>>4)
    const int row  = t & 15;
    const int slot = t >> 4;
    float topv[KNN];
    int   topi[KNN];
#pragma unroll
    for (int i = 0; i < KNN; ++i) { topv[i] = -3.0e38f; topi[i] = 0x7fffffff; }

    for (int chunk = 0; chunk < NPTS / 256; ++chunk) {
        const int chunkBase = chunk * 256;
        // 16 column tiles per chunk; 8 waves -> 2 tiles each (uniform EXEC)
#pragma unroll
        for (int tt = 0; tt < 2; ++tt) {
            const int tile    = wave * 2 + tt;
            const int colBase = chunkBase + tile * 16;
            v2f bfrag;
            bfrag.x = xs[kb * NPTS + colBase + mn];
            bfrag.y = xs[(kb + 1) * NPTS + colBase + mn];
            v8f c = {};
            v8f d = __builtin_amdgcn_wmma_f32_16x16x4_f32(
                false, afrag, false, bfrag, (short)0, c, false, false);
            const float xxc = xx[colBase + mn];
#pragma unroll
            for (int r = 0; r < 8; ++r) {
                const int m = r + hi * 8;
                dist[m * 264 + tile * 16 + mn] = 2.0f * d[r] - xxr[r] - xxc;
            }
        }
        __syncthreads();
        // selection update: scan 16 columns of this chunk
        for (int i = 0; i < 16; ++i) {
            const int   cc   = slot * 16 + i;
            const float v    = dist[row * 264 + cc];
            const int   gcol = chunkBase + cc;
            if (v > topv[KNN - 1]) {
                int p = KNN - 1;
                while (p > 0 && topv[p - 1] < v) {
                    topv[p] = topv[p - 1]; topi[p] = topi[p - 1]; --p;
                }
                topv[p] = v; topi[p] = gcol;
            }
        }
        __syncthreads();
    }

    // merge the 16 sorted lists per row
#pragma unroll
    for (int i = 0; i < KNN; ++i) {
        mVal[(row * 16 + slot) * KNN + i] = topv[i];
        mIdx[(row * 16 + slot) * KNN + i] = topi[i];
    }
    __syncthreads();
    if (t < 16) {
        int head[16];
#pragma unroll
        for (int s = 0; s < 16; ++s) head[s] = 0;
        for (int kk = 0; kk < KNN; ++kk) {
            float best = -3.4e38f; int bs = 0; int bidx = 0x7fffffff;
            for (int s = 0; s < 16; ++s) {
                if (head[s] < KNN) {
                    float v  = mVal[(t * 16 + s) * KNN + head[s]];
                    int   ix = mIdx[(t * 16 + s) * KNN + head[s]];
                    if (v > best || (v == best && ix < bidx)) {
                        best = v; bs = s; bidx = ix;
                    }
                }
            }
            idxOut[((size_t)bi * NPTS + rowBase + t) * KNN + kk] = bidx;
            head[bs]++;
        }
    }
}

// ---------------------------------------------------------------------------
// K2: per-channel sum / sum-of-squares of vector norms (BN pass 1).
// One thread per (b,n); LDS float atomics then one global atomic per block.
// ---------------------------------------------------------------------------
__global__ void stats_kernel(const float* __restrict__ x,
                             const float* __restrict__ Wf,
                             const int*   __restrict__ idx,
                             float* __restrict__ sums) {
    __shared__ float acc[128];
    const int t = threadIdx.x;
    if (t < 128) acc[t] = 0.0f;
    __syncthreads();

    const int gid = blockIdx.x * blockDim.x + t;     // (b,n)
    const int bi  = gid / NPTS;
    const int n   = gid % NPTS;
    const float* xb = x + (size_t)bi * 3 * NPTS;
    const float xi0 = xb[n], xi1 = xb[NPTS + n], xi2 = xb[2 * NPTS + n];

    float s1[NO], s2[NO];
#pragma unroll
    for (int o = 0; o < NO; ++o) { s1[o] = 0.0f; s2[o] = 0.0f; }

    for (int kk = 0; kk < KNN; ++kk) {
        const int j = idx[((size_t)bi * NPTS + n) * KNN + kk];
        const float e0 = xb[j] - xi0;
        const float e1 = xb[NPTS + j] - xi1;
        const float e2 = xb[2 * NPTS + j] - xi2;
#pragma unroll
        for (int o = 0; o < NO; ++o) {
            const float w0 = Wf[o * 2], w1 = Wf[o * 2 + 1];
            const float p0 = w0 * e0 + w1 * xi0;
            const float p1 = w0 * e1 + w1 * xi1;
            const float p2 = w0 * e2 + w1 * xi2;
            const float nr = sqrtf(p0 * p0 + p1 * p1 + p2 * p2) + VN_EPS;
            s1[o] += nr;
            s2[o] += nr * nr;
        }
    }
#pragma unroll
    for (int o = 0; o < NO; ++o) {
        atomicAdd(&acc[o], s1[o]);
        atomicAdd(&acc[64 + o], s2[o]);
    }
    __syncthreads();
    if (t < NO) {
        atomicAdd(&sums[t], acc[t]);
        atomicAdd(&sums[64 + t], acc[64 + t]);
    }
}

// ---------------------------------------------------------------------------
// K3: finalize BN stats -> mean, inv_std per channel
// ---------------------------------------------------------------------------
__global__ void finalize_kernel(const float* __restrict__ sums,
                                float* __restrict__ stats) {
    const int o = threadIdx.x;
    if (o < NO) {
        const float cnt  = (float)((size_t)NB * NPTS * KNN);
        const float mean = sums[o] / cnt;
        const float var  = sums[64 + o] / cnt - mean * mean;
        stats[o]      = mean;
        stats[64 + o] = rsqrtf(var + BN_EPS);
    }
}

// ---------------------------------------------------------------------------
// K4: fused VN-linear + BN + VN-leaky-ReLU + mean pool over k.
// One thread per (b,n,o). Recomputes p (2 weights/channel) instead of
// materializing the 82 MB intermediate.
// ---------------------------------------------------------------------------
__global__ void vn_kernel(const float* __restrict__ x,
                          const float* __restrict__ Wf,
                          const float* __restrict__ Wd,
                          const float* __restrict__ gamma,
                          const float* __restrict__ beta,
                          const int*   __restrict__ idx,
                          const float* __restrict__ stats,
                          float* __restrict__ out) {
    const int gid = blockIdx.x * blockDim.x + threadIdx.x;
    if (gid >= NB * NPTS * NO) return;
    const int o  = gid % NO;
    const int bn = gid / NO;
    const int n  = bn % NPTS;
    const int bi = bn / NPTS;

    const float* xb = x + (size_t)bi * 3 * NPTS;
    const float xi0 = xb[n], xi1 = xb[NPTS + n], xi2 = xb[2 * NPTS + n];
    const float wf0 = Wf[o * 2], wf1 = Wf[o * 2 + 1];
    const float wd0 = Wd[o * 2], wd1 = Wd[o * 2 + 1];
    const float gm = gamma[o], bt = beta[o];
    const float mean = stats[o], inv = stats[64 + o];

    float a0 = 0.0f, a1 = 0.0f, a2 = 0.0f;
    for (int kk = 0; kk < KNN; ++kk) {
        const int j = idx[(size_t)bn * KNN + kk];
        const float e0 = xb[j] - xi0;
        const float e1 = xb[NPTS + j] - xi1;
        const float e2 = xb[2 * NPTS + j] - xi2;
        float p0 = wf0 * e0 + wf1 * xi0;
        float p1 = wf0 * e1 + wf1 * xi1;
        float p2 = wf0 * e2 + wf1 * xi2;
        const float nr = sqrtf(p0 * p0 + p1 * p1 + p2 * p2) + VN_EPS;
        const float nb = gm * (nr - mean) * inv + bt;
        const float sc = nb / nr;
        p0 *= sc; p1 *= sc; p2 *= sc;
        const float d0 = wd0 * e0 + wd1 * xi0;
        const float d1 = wd0 * e1 + wd1 * xi1;
        const float d2 = wd0 * e2 + wd1 * xi2;
        const float dt = p0 * d0 + p1 * d1 + p2 * d2;
        const float dn = d0 * d0 + d1 * d1 + d2 * d2;
        float y0, y1, y2;
        if (dt >= 0.0f) {
            y0 = p0; y1 = p1; y2 = p2;
        } else {
            const float s = dt / (dn + VN_EPS);
            y0 = p0 - s * d0; y1 = p1 - s * d1; y2 = p2 - s * d2;
        }
        a0 += NEG_SLOPE * p0 + (1.0f - NEG_SLOPE) * y0;
        a1 += NEG_SLOPE * p1 + (1.0f - NEG_SLOPE) * y1;
        a2 += NEG_SLOPE * p2 + (1.0f - NEG_SLOPE) * y2;
    }
    const float invk = 1.0f / (float)KNN;
    const size_t base = (((size_t)bi * NO + o) * 3) * NPTS + n;
    out[base]            = a0 * invk;
    out[base + NPTS]     = a1 * invk;
    out[base + 2 * NPTS] = a2 * invk;
}

// ---------------------------------------------------------------------------
extern "C" void kernel_launch(void* const* d_in, const int* in_sizes, int n_in,
                              void* d_out, int out_size, void* d_ws, size_t ws_size,
                              hipStream_t stream) {
    const float* x     = (const float*)d_in[0];
    const float* Wf    = (const float*)d_in[1];
    const float* Wd    = (const float*)d_in[2];
    const float* gamma = (const float*)d_in[3];
    const float* beta  = (const float*)d_in[4];
    float* out = (float*)d_out;

    // workspace layout
    int*   idx   = (int*)d_ws;                                       // [B][N][KNN]
    float* sums  = (float*)((char*)d_ws +
                            (size_t)NB * NPTS * KNN * sizeof(int));  // [128]
    float* stats = sums + 128;                                       // [128]

    zero_stats_kernel<<<1, 128, 0, stream>>>(sums);

    const size_t shmem = (4 * NPTS + NPTS + 16 * 264 + 16 * 16 * KNN) * sizeof(float)
                       + (size_t)16 * 16 * KNN * sizeof(int);        // ~97 KB
    knn_kernel<<<NB * (NPTS / 16), 256, shmem, stream>>>(x, idx);

    stats_kernel<<<(NB * NPTS) / 256, 256, 0, stream>>>(x, Wf, idx, sums);
    finalize_kernel<<<1, 32, 0, stream>>>(sums, stats);

    const int total = NB * NPTS * NO;
    vn_kernel<<<(total + 255) / 256, 256, 0, stream>>>(
        x, Wf, Wd, gamma, beta, idx, stats, out);
}